// ResGATLayer_76596446757096
// MI455X (gfx1250) — compile-verified
//
#include <hip/hip_runtime.h>
#include <math.h>

#define N_NODES 50000
#define N_EDGES 800000
#define D_FEAT  128
#define HEADS   8
#define HEAD_DIM 16

typedef __attribute__((ext_vector_type(2))) float v2f;
typedef __attribute__((ext_vector_type(8))) float v8f;

// ---------------------------------------------------------------------------
// Kernel 0: init accumulators (runs every call; no cross-call state)
// ---------------------------------------------------------------------------
__global__ void init_ws(float* __restrict__ out_acc,
                        float* __restrict__ nmax,
                        float* __restrict__ denom) {
    int i = blockIdx.x * blockDim.x + threadIdx.x;
    if (i < N_NODES * D_FEAT) out_acc[i] = 0.0f;
    if (i < N_NODES * HEADS) {
        nmax[i]  = __int_as_float(0xFF800000); // -inf
        denom[i] = 0.0f;
    }
}

// ---------------------------------------------------------------------------
// Kernel 1: h = x @ W via V_WMMA_F32_16X16X4_F32 (f32 exact path)
// grid = 3125 blocks (M tiles of 16), block = 256 (8 waves, one per N tile)
// ---------------------------------------------------------------------------
__global__ void gemm_wmma(const float* __restrict__ x,
                          const float* __restrict__ W,
                          float* __restrict__ h) {
    __shared__ float As[16 * 128]; // 8 KB x-tile
    const int t  = threadIdx.x;
    const int m0 = blockIdx.x * 16;

    // Coalesced load of 16x128 x-tile (512 float4s by 256 threads)
    {
        const float4* xg  = (const float4*)(x + (size_t)m0 * D_FEAT);
        float4*       as4 = (float4*)As;
        for (int i = t; i < 16 * 32; i += 256) as4[i] = xg[i];
    }
    __syncthreads();

    const int wave = t >> 5;        // N-tile index (0..7)
    const int lane = t & 31;
    const int half = lane >> 4;     // 0: lanes 0-15, 1: lanes 16-31
    const int lrow = lane & 15;
    const int col0 = wave * 16;

    v8f c = {};
    #pragma unroll 4
    for (int k0 = 0; k0 < 128; k0 += 4) {
        const int k = k0 + 2 * half;
        v2f a, b;
        // A 16x4 layout: lane = M; VGPR0 = K (lanes0-15) / K+2 (lanes16-31)
        a[0] = As[lrow * 128 + k];
        a[1] = As[lrow * 128 + k + 1];
        // B 4x16 layout: N striped across lanes within a VGPR
        b[0] = W[(size_t)k       * 128 + col0 + lrow];
        b[1] = W[(size_t)(k + 1) * 128 + col0 + lrow];
        c = __builtin_amdgcn_wmma_f32_16x16x4_f32(false, a, false, b,
                                                  (short)0, c, false, false);
    }
    // C/D layout: VGPR r -> row r (lanes 0-15) / row r+8 (lanes 16-31)
    #pragma unroll
    for (int r = 0; r < 8; ++r)
        h[(size_t)(m0 + r + 8 * half) * D_FEAT + col0 + lrow] = c[r];
}

// ---------------------------------------------------------------------------
// Kernel 2: per-(node,head) attention logit components
// ---------------------------------------------------------------------------
__global__ void attn_node(const float* __restrict__ h,
                          const float* __restrict__ att_src,
                          const float* __restrict__ att_dst,
                          float* __restrict__ a_src,
                          float* __restrict__ a_dst) {
    int i = blockIdx.x * blockDim.x + threadIdx.x;
    if (i >= N_NODES * HEADS) return;
    const int head = i & (HEADS - 1);
    // i = n*8 + head  ->  h offset = n*128 + head*16 = i*16
    const float4* hv = (const float4*)(h + (size_t)i * HEAD_DIM);
    const float4* as = (const float4*)(att_src + head * HEAD_DIM);
    const float4* ad = (const float4*)(att_dst + head * HEAD_DIM);
    float s = 0.0f, d = 0.0f;
    #pragma unroll
    for (int j = 0; j < 4; ++j) {
        float4 hx = hv[j], av = as[j], dv = ad[j];
        s += hx.x * av.x + hx.y * av.y + hx.z * av.z + hx.w * av.w;
        d += hx.x * dv.x + hx.y * dv.y + hx.z * dv.z + hx.w * dv.w;
    }
    a_src[i] = s;
    a_dst[i] = d;
}

// float atomic-max via signed-max / unsigned-min bit trick (order-safe)
__device__ __forceinline__ void atomicMaxFloat(float* addr, float val) {
    if (val >= 0.0f) atomicMax((int*)addr, __float_as_int(val));
    else             atomicMin((unsigned int*)addr, (unsigned int)__float_as_int(val));
}

// ---------------------------------------------------------------------------
// Kernel 3: per-(edge,head) logits + segment max
// ---------------------------------------------------------------------------
__global__ void edge_logits(const int* __restrict__ ei,
                            const float* __restrict__ a_src,
                            const float* __restrict__ a_dst,
                            float* __restrict__ e_buf,
                            float* __restrict__ nmax) {
    int i = blockIdx.x * blockDim.x + threadIdx.x;
    if (i >= N_EDGES * HEADS) return;
    const int e    = i >> 3;
    const int head = i & 7;
    const int src = ei[e];
    const int dst = ei[N_EDGES + e];
    float v = a_src[src * HEADS + head] + a_dst[dst * HEADS + head];
    v = (v > 0.0f) ? v : 0.2f * v;   // LeakyReLU(0.2)
    e_buf[i] = v;
    atomicMaxFloat(&nmax[dst * HEADS + head], v);
}

// ---------------------------------------------------------------------------
// Kernel 4: p = exp(e - max[dst]); accumulate denominators
// ---------------------------------------------------------------------------
__global__ void edge_softmax(const int* __restrict__ ei,
                             float* __restrict__ e_buf,
                             const float* __restrict__ nmax,
                             float* __restrict__ denom) {
    int i = blockIdx.x * blockDim.x + threadIdx.x;
    if (i >= N_EDGES * HEADS) return;
    const int e    = i >> 3;
    const int head = i & 7;
    const int dst  = ei[N_EDGES + e];
    float p = __expf(e_buf[i] - nmax[dst * HEADS + head]);
    e_buf[i] = p;
    atomicAdd(&denom[dst * HEADS + head], p);
}

// ---------------------------------------------------------------------------
// Kernel 5: message scatter. One wave32 per edge: 32 lanes x float4 reads the
// full 512B row of h[src] in one coalesced transaction.
// ---------------------------------------------------------------------------
__global__ void edge_scatter(const int* __restrict__ ei,
                             const float* __restrict__ h,
                             const float* __restrict__ e_buf,
                             const float* __restrict__ denom,
                             float* __restrict__ out_acc) {
    int gid  = blockIdx.x * blockDim.x + threadIdx.x;
    int edge = gid >> 5;
    int lane = gid & 31;
    if (edge >= N_EDGES) return;
    const int src  = ei[edge];
    const int dst  = ei[N_EDGES + edge];
    const int head = lane >> 2;                 // 4 lanes per head
    float p = e_buf[edge * HEADS + head];
    float d = denom[dst * HEADS + head];
    float alpha = p / (d + 1e-16f);
    float4 hv = ((const float4*)(h + (size_t)src * D_FEAT))[lane];
    float* o = out_acc + (size_t)dst * D_FEAT + lane * 4;
    atomicAdd(o + 0, hv.x * alpha);
    atomicAdd(o + 1, hv.y * alpha);
    atomicAdd(o + 2, hv.z * alpha);
    atomicAdd(o + 3, hv.w * alpha);
}

// ---------------------------------------------------------------------------
// Kernel 6: +bias, +residual, LayerNorm. One wave32 per node (4 f32/lane).
// ---------------------------------------------------------------------------
__global__ void finalize_ln(const float* __restrict__ x,
                            const float* __restrict__ bias,
                            const float* __restrict__ gamma,
                            const float* __restrict__ beta,
                            float* __restrict__ out) {
    int gid  = blockIdx.x * blockDim.x + threadIdx.x;
    int node = gid >> 5;
    int lane = gid & 31;
    if (node >= N_NODES) return;

    float4 v  = ((float4*)out)[(size_t)node * 32 + lane];
    float4 bv = ((const float4*)bias)[lane];
    float4 xv = ((const float4*)x)[(size_t)node * 32 + lane];
    v.x += bv.x + xv.x; v.y += bv.y + xv.y;
    v.z += bv.z + xv.z; v.w += bv.w + xv.w;

    float s = v.x + v.y + v.z + v.w;
    float q = v.x * v.x + v.y * v.y + v.z * v.z + v.w * v.w;
    #pragma unroll
    for (int m = 16; m >= 1; m >>= 1) {
        s += __shfl_xor(s, m, 32);
        q += __shfl_xor(q, m, 32);
    }
    float mean = s * (1.0f / D_FEAT);
    float var  = q * (1.0f / D_FEAT) - mean * mean;
    float rstd = rsqrtf(var + 1e-5f);

    float4 gv = ((const float4*)gamma)[lane];
    float4 be = ((const float4*)beta)[lane];
    float4 r;
    r.x = (v.x - mean) * rstd * gv.x + be.x;
    r.y = (v.y - mean) * rstd * gv.y + be.y;
    r.z = (v.z - mean) * rstd * gv.z + be.z;
    r.w = (v.w - mean) * rstd * gv.w + be.w;
    ((float4*)out)[(size_t)node * 32 + lane] = r;
}

// ---------------------------------------------------------------------------
extern "C" void kernel_launch(void* const* d_in, const int* in_sizes, int n_in,
                              void* d_out, int out_size, void* d_ws, size_t ws_size,
                              hipStream_t stream) {
    const float* x       = (const float*)d_in[0];
    const int*   ei      = (const int*)  d_in[1];
    const float* W       = (const float*)d_in[2];
    const float* att_src = (const float*)d_in[3];
    const float* att_dst = (const float*)d_in[4];
    const float* bias    = (const float*)d_in[5];
    const float* gamma   = (const float*)d_in[6];
    const float* beta    = (const float*)d_in[7];
    float* out = (float*)d_out;

    // workspace layout
    char* ws = (char*)d_ws;
    float* h      = (float*)ws;                                   ws += (size_t)N_NODES * D_FEAT * 4;
    float* a_src  = (float*)ws;                                   ws += (size_t)N_NODES * HEADS * 4;
    float* a_dst  = (float*)ws;                                   ws += (size_t)N_NODES * HEADS * 4;
    float* nmax   = (float*)ws;                                   ws += (size_t)N_NODES * HEADS * 4;
    float* denom  = (float*)ws;                                   ws += (size_t)N_NODES * HEADS * 4;
    float* e_buf  = (float*)ws;                                   ws += (size_t)N_EDGES * HEADS * 4;

    // 0) init accumulators (every call)
    {
        int total = N_NODES * D_FEAT;
        init_ws<<<(total + 255) / 256, 256, 0, stream>>>(out, nmax, denom);
    }
    // 1) h = x @ W  (WMMA f32)
    gemm_wmma<<<N_NODES / 16, 256, 0, stream>>>(x, W, h);
    // 2) node attention components
    {
        int total = N_NODES * HEADS;
        attn_node<<<(total + 255) / 256, 256, 0, stream>>>(h, att_src, att_dst, a_src, a_dst);
    }
    // 3) edge logits + segment max
    {
        int total = N_EDGES * HEADS;
        edge_logits<<<(total + 255) / 256, 256, 0, stream>>>(ei, a_src, a_dst, e_buf, nmax);
    }
    // 4) softmax numerator + denominators
    {
        int total = N_EDGES * HEADS;
        edge_softmax<<<(total + 255) / 256, 256, 0, stream>>>(ei, e_buf, nmax, denom);
    }
    // 5) weighted scatter (wave per edge)
    {
        long long total = (long long)N_EDGES * 32;
        edge_scatter<<<(int)(total / 256), 256, 0, stream>>>(ei, h, e_buf, denom, out);
    }
    // 6) bias + residual + LayerNorm (wave per node)
    {
        long long total = (long long)N_NODES * 32;
        finalize_ln<<<(int)(total / 256), 256, 0, stream>>>(x, bias, gamma, beta, out);
    }
}